// LossWithBeliveMaps_68410239091404
// MI455X (gfx1250) — compile-verified
//
#include <hip/hip_runtime.h>

typedef __attribute__((ext_vector_type(2))) float v2f;
typedef __attribute__((ext_vector_type(8))) float v8f;

#define BB   16
#define HH   1024
#define WW   1024
#define NKP  100
#define KS   9
#define KR   4
#define ROWS 8            // stripe height; LDS tile = 8*1024*4B = 32 KB
#define TPB  256

__global__ void belive_zero_out(float* out) { out[0] = 0.0f; }

__global__ __launch_bounds__(TPB)
void belive_loss_kernel(const float* __restrict__ pred,
                        const int*   __restrict__ coords,
                        const int*   __restrict__ labels,
                        const float* __restrict__ gk,
                        float*       __restrict__ out)
{
    __shared__ __align__(16) float s_tile[ROWS * WW];   // belief-map stripe
    __shared__ int   s_kx[NKP];
    __shared__ int   s_ky[NKP];
    __shared__ int   s_nc;
    __shared__ float s_part[TPB / 32];

    const int tilesPerB = HH / ROWS;                    // 128
    const int b   = blockIdx.x / tilesPerB;
    const int r0  = (blockIdx.x % tilesPerB) * ROWS;
    const int tid = threadIdx.x;

    const float4* p4 = (const float4*)(pred + ((size_t)b * HH + (size_t)r0) * WW);

    // Warm the cache for this thread's 8 rows while LDS phases run.
    #pragma unroll
    for (int k = 0; k < ROWS; ++k)
        __builtin_prefetch(&p4[tid + (k << 8)], 0, 3);

    // ---- Phase 0: zero belief tile ----
    if (tid == 0) s_nc = 0;
    {
        float4* t4 = (float4*)s_tile;
        #pragma unroll
        for (int i = tid; i < ROWS * WW / 4; i += TPB)
            t4[i] = make_float4(0.f, 0.f, 0.f, 0.f);
    }
    __syncthreads();

    // ---- Phase 1: parallel candidate build with global first-occurrence dedup.
    // scatter .at[].set(1.0) semantics: identical (x,y) counts once; duplicates
    // always share the same stripes, so first-occurrence dedup is exact.
    if (tid < NKP) {
        int ch = labels[b * NKP + tid] - 1;
        int x  = coords[(b * NKP + tid) * 2 + 0];
        int y  = coords[(b * NKP + tid) * 2 + 1];
        bool keep = (ch == 0) && (y >= r0 - KR) && (y < r0 + ROWS + KR);
        if (keep) {
            for (int j = 0; j < tid; ++j) {
                if (labels[b * NKP + j] == 1 &&
                    coords[(b * NKP + j) * 2 + 0] == x &&
                    coords[(b * NKP + j) * 2 + 1] == y) { keep = false; break; }
            }
        }
        if (keep) {
            int slot = atomicAdd(&s_nc, 1);
            s_kx[slot] = x;
            s_ky[slot] = y;
        }
    }
    __syncthreads();

    // ---- Phase 2: splat 9x9 gaussian windows into the tile (sums overlap).
    // Cross-correlation of a delta at m: out[m+d] += K[d+4]; border clipped = conv padding.
    {
        const int total = s_nc * (KS * KS);
        for (int idx = tid; idx < total; idx += TPB) {
            int kp = idx / (KS * KS);
            int c  = idx - kp * (KS * KS);
            int dy = c / KS - KR;
            int dx = c - (c / KS) * KS - KR;
            int y  = s_ky[kp] + dy;
            int x  = s_kx[kp] + dx;
            if (y >= r0 && y < r0 + ROWS && (unsigned)x < (unsigned)WW)
                atomicAdd(&s_tile[(y - r0) * WW + x], gk[c]);
        }
    }
    __syncthreads();

    // ---- Phase 3: branch-free stream.  d = pred - belief; accumulate d^2 on
    // the WMMA accumulator diagonal (same regs as A and B => B = A^T by the
    // 16x4 / 4x16 lane layouts, so D = A*A^T + C).
    const float4* t4 = (const float4*)s_tile;
    v8f cacc0 = {0.f, 0.f, 0.f, 0.f, 0.f, 0.f, 0.f, 0.f};
    v8f cacc1 = {0.f, 0.f, 0.f, 0.f, 0.f, 0.f, 0.f, 0.f};

    #pragma unroll
    for (int k = 0; k < ROWS; ++k) {
        float4 pv = p4[tid + (k << 8)];
        float4 tv = t4[tid + (k << 8)];
        v2f a0 = { pv.x - tv.x, pv.y - tv.y };
        v2f a1 = { pv.z - tv.z, pv.w - tv.w };
        cacc0 = __builtin_amdgcn_wmma_f32_16x16x4_f32(
            false, a0, false, a0, (short)0, cacc0, false, false);
        cacc1 = __builtin_amdgcn_wmma_f32_16x16x4_f32(
            false, a1, false, a1, (short)0, cacc1, false, false);
    }

    v8f cacc = cacc0 + cacc1;

    // Diagonal of 16x16 f32 C/D tile: lanes 0-7 at VGPR=lane; lanes 24-31 at VGPR=lane-24.
    const int lane = tid & 31;
    float diag = 0.f;
    if (lane < 8)        diag = cacc[lane];
    else if (lane >= 24) diag = cacc[lane - 24];

    for (int off = 16; off > 0; off >>= 1)
        diag += __shfl_xor(diag, off, 32);

    if (lane == 0) s_part[tid >> 5] = diag;
    __syncthreads();

    if (tid == 0) {
        float t = 0.f;
        for (int w = 0; w < TPB / 32; ++w) t += s_part[w];
        atomicAdd(out, t * (1.0f / ((float)BB * (float)HH * (float)WW)));
    }
}

extern "C" void kernel_launch(void* const* d_in, const int* in_sizes, int n_in,
                              void* d_out, int out_size, void* d_ws, size_t ws_size,
                              hipStream_t stream) {
    (void)in_sizes; (void)n_in; (void)out_size; (void)d_ws; (void)ws_size;
    const float* pred   = (const float*)d_in[0];
    const int*   coords = (const int*)d_in[1];
    const int*   labels = (const int*)d_in[2];
    const float* gk     = (const float*)d_in[3];
    float* out = (float*)d_out;

    belive_zero_out<<<1, 1, 0, stream>>>(out);
    belive_loss_kernel<<<dim3(BB * (HH / ROWS)), dim3(TPB), 0, stream>>>(
        pred, coords, labels, gk, out);
}